// GATv2_46308337385568
// MI455X (gfx1250) — compile-verified
//
#include <hip/hip_runtime.h>
#include <hip/hip_bf16.h>

#define H 4
#define C 32
#define HC 128
#define NEG_SLOPE 0.2f
#define EPB 16          // edges per block in edge kernels

typedef float v2f __attribute__((ext_vector_type(2)));
typedef float v8f __attribute__((ext_vector_type(8)));

// ---------- helpers ----------
__device__ __forceinline__ unsigned fkey(float f) {
    unsigned u = __float_as_uint(f);
    return (u & 0x80000000u) ? ~u : (u | 0x80000000u);
}
__device__ __forceinline__ float finv(unsigned k) {
    unsigned u = (k & 0x80000000u) ? (k & 0x7fffffffu) : ~k;
    return __uint_as_float(u);
}

// ---------- self-loop edge_attr mean (fill_value='mean') ----------
__global__ void selfloop_accum_kernel(const int* __restrict__ dst,
                                      const float* __restrict__ ea,
                                      float* __restrict__ cnt,
                                      float* __restrict__ easum, int E) {
    int e = blockIdx.x * blockDim.x + threadIdx.x;
    if (e >= E) return;
    int d = dst[e];
    unsafeAtomicAdd(cnt + d, 1.0f);
#pragma unroll
    for (int k = 0; k < 4; ++k)
        unsafeAtomicAdd(easum + (size_t)d * 4 + k, ea[(size_t)e * 4 + k]);
}

__global__ void selfloop_mean_kernel(const float* __restrict__ cnt,
                                     float* __restrict__ eamean, int N) {
    int i = blockIdx.x * blockDim.x + threadIdx.x;
    if (i >= N * 4) return;
    float c = cnt[i >> 2];
    eamean[i] = eamean[i] / (c > 1.0f ? c : 1.0f);
}

// ---------- node transform GEMM: out[N,128] = h[N,FIN] @ W[FIN,128] + b ----------
// 128 threads = 4 waves; each wave computes a 16x128 tile with 8 WMMA accumulators.
template <int FIN>
__global__ void gemm_node_kernel(const float* __restrict__ h,
                                 const float* __restrict__ W,
                                 const float* __restrict__ bias,
                                 float* __restrict__ out, int N) {
    __shared__ float sW[FIN * HC];
    __shared__ float sB[HC];
    int tid = threadIdx.x;                       // 0..127
    for (int i = tid; i < FIN * HC; i += 128) sW[i] = W[i];
    sB[tid] = bias[tid];
    __syncthreads();

    int wave = tid >> 5;
    int lane = tid & 31;
    int half = lane >> 4;                        // 0: lanes 0-15, 1: lanes 16-31
    int l16  = lane & 15;
    long rowbase = ((long)blockIdx.x * 4 + wave) * 16;
    long arow  = rowbase + l16;
    long arowc = arow < N ? arow : (long)(N - 1);  // clamp: EXEC must stay all-1s

    v8f acc[8];
#pragma unroll
    for (int nt = 0; nt < 8; ++nt) acc[nt] = {};

#pragma unroll
    for (int k0 = 0; k0 < FIN; k0 += 4) {
        int ka = k0 + half * 2;                  // ISA 16x4 fp32 A layout
        v2f a;
        a.x = h[arowc * FIN + ka];
        a.y = h[arowc * FIN + ka + 1];
#pragma unroll
        for (int nt = 0; nt < 8; ++nt) {
            int n = nt * 16 + l16;               // mirrored 4x16 B layout
            v2f b;
            b.x = sW[ka * HC + n];
            b.y = sW[(ka + 1) * HC + n];
            acc[nt] = __builtin_amdgcn_wmma_f32_16x16x4_f32(
                false, a, false, b, (short)0, acc[nt], false, false);
        }
    }

    // ---- epilogue ----
    // C/D layout: lanes 0-15 -> M = r, lanes 16-31 -> M = 8 + r.
    long mbase = rowbase + half * 8;
    // Wave-uniform fast path: entire 16-row tile in range -> unguarded stores
    // with constant immediate offsets (r * 512B). readfirstlane forces a
    // single scalar branch instead of per-element EXEC juggling.
    int full = (int)((rowbase + 16) <= (long)N);
    if (__builtin_amdgcn_readfirstlane(full)) {
#pragma unroll
        for (int nt = 0; nt < 8; ++nt) {
            int n = nt * 16 + l16;
            float bz = sB[n];
            float* po = out + mbase * HC + n;
#pragma unroll
            for (int r = 0; r < 8; ++r)
                po[r * HC] = acc[nt][r] + bz;
        }
    } else {
#pragma unroll
        for (int nt = 0; nt < 8; ++nt) {
            int n = nt * 16 + l16;
            float bz = sB[n];
            float* po = out + mbase * HC + n;
#pragma unroll
            for (int r = 0; r < 8; ++r)
                if (mbase + r < (long)N) po[r * HC] = acc[nt][r] + bz;
        }
    }
}

// ---------- edge pass 1: logits + segment max ----------
// 128 threads/block: wave w handles head w, lane c handles channel c.
__global__ void edge_logits_kernel(const int* __restrict__ src,
                                   const int* __restrict__ dst,
                                   const float* __restrict__ ea,
                                   const float* __restrict__ eamean,
                                   const float* __restrict__ xl,
                                   const float* __restrict__ xr,
                                   const float* __restrict__ We,    // 4 x 128
                                   const float* __restrict__ att,   // 128
                                   float* __restrict__ logits,      // Etot*4
                                   unsigned* __restrict__ lmax,     // N*4 keys
                                   int E, int Etot) {
    __shared__ float sWe[4 * HC];
    __shared__ float sAtt[HC];
    int tid = threadIdx.x;
    for (int i = tid; i < 4 * HC; i += 128) sWe[i] = We[i];
    sAtt[tid] = att[tid];
    __syncthreads();

    int hd = tid >> 5, c = tid & 31, ch = tid;
    long e0 = (long)blockIdx.x * EPB;
    for (int i = 0; i < EPB; ++i) {
        long e = e0 + i;
        if (e >= Etot) return;
        int s, d; const float* eap;
        if (e < E) { s = src[e]; d = dst[e]; eap = ea + e * 4; }
        else       { int n = (int)(e - E); s = n; d = n; eap = eamean + (size_t)n * 4; }

        float ev = eap[0] * sWe[ch] + eap[1] * sWe[HC + ch] +
                   eap[2] * sWe[2 * HC + ch] + eap[3] * sWe[3 * HC + ch];
        float m = xl[(size_t)s * HC + ch] + xr[(size_t)d * HC + ch] + ev;
        m = m > 0.0f ? m : NEG_SLOPE * m;        // leaky_relu
        float p = m * sAtt[ch];
#pragma unroll
        for (int off = 16; off; off >>= 1) p += __shfl_xor(p, off, 32);
        if (c == 0) {
            logits[e * 4 + hd] = p;
            atomicMax(lmax + (size_t)d * 4 + hd, fkey(p));
        }
    }
}

// ---------- edge pass 2: exp + denom ----------
__global__ void edge_exp_kernel(const int* __restrict__ dst,
                                const float* __restrict__ logits,
                                const unsigned* __restrict__ lmax,
                                float* __restrict__ exbuf,
                                float* __restrict__ denom,
                                int E, int Etot) {
    long idx = (long)blockIdx.x * blockDim.x + threadIdx.x;
    if (idx >= (long)Etot * 4) return;
    long e = idx >> 2;
    int hd = (int)(idx & 3);
    int d = (e < E) ? dst[e] : (int)(e - E);
    float lm = finv(lmax[(size_t)d * 4 + hd]);
    float ex = __expf(logits[idx] - lm);
    exbuf[idx] = ex;
    unsafeAtomicAdd(denom + (size_t)d * 4 + hd, ex);
}

// ---------- edge pass 3: alpha-weighted aggregation ----------
__global__ void edge_agg_kernel(const int* __restrict__ src,
                                const int* __restrict__ dst,
                                const float* __restrict__ exbuf,
                                const float* __restrict__ denom,
                                const float* __restrict__ xl,
                                float* __restrict__ agg,   // N*128
                                int E, int Etot) {
    int tid = threadIdx.x;                 // 128
    int hd = tid >> 5, ch = tid;
    long e0 = (long)blockIdx.x * EPB;
    for (int i = 0; i < EPB; ++i) {
        long e = e0 + i;
        if (e >= Etot) return;
        int s, d;
        if (e < E) { s = src[e]; d = dst[e]; }
        else       { int n = (int)(e - E); s = n; d = n; }
        float alpha = exbuf[e * 4 + hd] / (denom[(size_t)d * 4 + hd] + 1e-16f);
        float val = xl[(size_t)s * HC + ch] * alpha;
        unsafeAtomicAdd(agg + (size_t)d * HC + ch, val);
    }
}

// ---------- head mean + bias (+ELU) ----------
__global__ void finalize_kernel(const float* __restrict__ agg,
                                const float* __restrict__ bias,
                                float* __restrict__ out, int N, int do_elu) {
    long i = (long)blockIdx.x * blockDim.x + threadIdx.x;
    if (i >= (long)N * C) return;
    long n = i >> 5;
    int c = (int)(i & 31);
    const float* a = agg + n * HC;
    float v = 0.25f * (a[c] + a[C + c] + a[2 * C + c] + a[3 * C + c]) + bias[c];
    if (do_elu) v = v > 0.0f ? v : (__expf(v) - 1.0f);
    out[i] = v;
}

__global__ void mask_kernel(const float* __restrict__ x, int* __restrict__ mask, int N) {
    int n = blockIdx.x * blockDim.x + threadIdx.x;
    if (n < N) mask[n] = (x[(size_t)n * 4 + 3] != 0.0f) ? 1 : 0;
}

// ---------- launch ----------
extern "C" void kernel_launch(void* const* d_in, const int* in_sizes, int n_in,
                              void* d_out, int out_size, void* d_ws, size_t ws_size,
                              hipStream_t stream) {
    const float* x   = (const float*)d_in[0];
    const int*   ei  = (const int*)d_in[1];
    const float* ea  = (const float*)d_in[2];
    const int N = in_sizes[0] / 4;
    const int E = in_sizes[2] / 4;
    const int Etot = E + N;                    // edges + self loops
    const int* src = ei;
    const int* dst = ei + E;

    // workspace carve (all fp32-sized)
    float* ws = (float*)d_ws;
    float*    xl     = ws;                      ws += (size_t)N * HC;
    float*    xr     = ws;                      ws += (size_t)N * HC;
    float*    agg    = ws;                      ws += (size_t)N * HC;
    float*    hbuf   = ws;                      ws += (size_t)N * C;
    float*    logits = ws;                      ws += (size_t)Etot * 4;
    float*    exbuf  = ws;                      ws += (size_t)Etot * 4;
    unsigned* lmax   = (unsigned*)ws;           ws += (size_t)N * 4;
    float*    denom  = ws;                      ws += (size_t)N * 4;
    float*    eamean = ws;                      ws += (size_t)N * 4;
    float*    cnt    = ws;                      ws += (size_t)N;

    auto cdiv = [](long a, long b) { return (unsigned)((a + b - 1) / b); };

    // self-loop edge_attr mean
    hipMemsetAsync(eamean, 0, (size_t)N * 4 * sizeof(float), stream);
    hipMemsetAsync(cnt,    0, (size_t)N * sizeof(float), stream);
    selfloop_accum_kernel<<<cdiv(E, 256), 256, 0, stream>>>(dst, ea, cnt, eamean, E);
    selfloop_mean_kernel<<<cdiv((long)N * 4, 256), 256, 0, stream>>>(cnt, eamean, N);

    const float* hin = x;
    for (int L = 0; L < 4; ++L) {
        const float* Wl   = (const float*)d_in[4 + L * 7 + 0];
        const float* bl   = (const float*)d_in[4 + L * 7 + 1];
        const float* Wr   = (const float*)d_in[4 + L * 7 + 2];
        const float* br   = (const float*)d_in[4 + L * 7 + 3];
        const float* We   = (const float*)d_in[4 + L * 7 + 4];
        const float* att  = (const float*)d_in[4 + L * 7 + 5];
        const float* bias = (const float*)d_in[4 + L * 7 + 6];

        unsigned gblocks = cdiv(N, 64);        // 4 waves x 16 rows per block
        if (L == 0) {
            gemm_node_kernel<4><<<gblocks, 128, 0, stream>>>(hin, Wl, bl, xl, N);
            gemm_node_kernel<4><<<gblocks, 128, 0, stream>>>(hin, Wr, br, xr, N);
        } else {
            gemm_node_kernel<32><<<gblocks, 128, 0, stream>>>(hin, Wl, bl, xl, N);
            gemm_node_kernel<32><<<gblocks, 128, 0, stream>>>(hin, Wr, br, xr, N);
        }

        hipMemsetAsync(lmax,  0, (size_t)N * 4 * sizeof(unsigned), stream);
        hipMemsetAsync(denom, 0, (size_t)N * 4 * sizeof(float), stream);
        hipMemsetAsync(agg,   0, (size_t)N * HC * sizeof(float), stream);

        edge_logits_kernel<<<cdiv(Etot, EPB), 128, 0, stream>>>(
            src, dst, ea, eamean, xl, xr, We, att, logits, lmax, E, Etot);
        edge_exp_kernel<<<cdiv((long)Etot * 4, 256), 256, 0, stream>>>(
            dst, logits, lmax, exbuf, denom, E, Etot);
        edge_agg_kernel<<<cdiv(Etot, EPB), 128, 0, stream>>>(
            src, dst, exbuf, denom, xl, agg, E, Etot);

        float* hout = (L == 3) ? (float*)d_out : hbuf;
        finalize_kernel<<<cdiv((long)N * C, 256), 256, 0, stream>>>(
            agg, bias, hout, N, (L < 3) ? 1 : 0);
        hin = hbuf;
    }

    // mask output: (x[:,3] != 0), int32, after N*32 floats of encoded output
    int* maskout = (int*)d_out + (size_t)N * C;
    mask_kernel<<<cdiv(N, 256), 256, 0, stream>>>(x, maskout, N);
}